// simple_attn_5471788335757
// MI455X (gfx1250) — compile-verified
//
#include <hip/hip_runtime.h>
#include <hip/hip_bf16.h>
#include <math.h>

// Problem constants (from reference setup_inputs)
#define BB   4
#define NN   32768
#define CC   256
#define HH   8
#define HC   32
#define EPSF 1e-5f
#define INVN (1.0f / 32768.0f)

#define SPLIT   16          // blocks per (b,h)
#define CHUNK   128         // rows staged in LDS per iteration (pass 1)
#define PAD     40          // padded LDS row stride (floats): 40%64 banks -> conflict-free halves

typedef __attribute__((ext_vector_type(2))) float v2f;
typedef __attribute__((ext_vector_type(8))) float v8f;

__device__ __forceinline__ v8f wmma_f32_4(v2f a, v2f b, v8f c) {
    // D = A(16x4 f32) x B(4x16 f32) + C(16x16 f32)
    return __builtin_amdgcn_wmma_f32_16x16x4_f32(
        /*neg_a=*/false, a, /*neg_b=*/false, b,
        /*c_mod=*/(short)0, c, /*reuse_a=*/false, /*reuse_b=*/false);
}

// ---------------------------------------------------------------------------
// Kernel 0: zero the kv scratch (B*H*32*32 floats) in d_ws
// ---------------------------------------------------------------------------
__global__ void la_zero_kv(float* __restrict__ kv) {
    int i = blockIdx.x * blockDim.x + threadIdx.x;
    if (i < BB * HH * HC * HC) kv[i] = 0.0f;
}

// ---------------------------------------------------------------------------
// Kernel 1: LN(k), LN(v) + kv += k^T v  (per (b,h), over an N-split)
// grid = B*H*SPLIT blocks of 256 threads (8 waves)
// ---------------------------------------------------------------------------
__global__ __launch_bounds__(256) void la_pass1(
    const float* __restrict__ x,
    const float* __restrict__ wk, const float* __restrict__ bk,
    const float* __restrict__ wv, const float* __restrict__ bv,
    float* __restrict__ kv)
{
    __shared__ float kn_lds[CHUNK * PAD];
    __shared__ float vn_lds[CHUNK * PAD];

    const int split = blockIdx.x % SPLIT;
    const int bh    = blockIdx.x / SPLIT;
    const int h     = bh % HH;
    const int b     = bh / HH;

    const int tid  = threadIdx.x;
    const int wave = tid >> 5;
    const int lane = tid & 31;

    // ---- LN loader mapping: 8 lanes per row, float4 per lane ----
    const int dgrp = lane & 7;       // which float4 within the 32-float row
    const int rsub = lane >> 3;      // row within the warp's group of 4
    const int d0   = dgrp * 4;

    const float4 w_k = *(const float4*)(wk + h * HC + d0);
    const float4 b_k = *(const float4*)(bk + h * HC + d0);
    const float4 w_v = *(const float4*)(wv + h * HC + d0);
    const float4 b_v = *(const float4*)(bv + h * HC + d0);

    // ---- WMMA operand mapping (16x4 A / 4x16 B, f32) ----
    const int m   = lane & 15;          // A: M index / B: N index
    const int k2  = (lane >> 4) * 2;    // K sub-offset for this half-wave

    // 2x2 tiles of the 32x32 kv accumulator: [di*2 + ei]
    v8f acc[4] = {};

    const int rows_per_block = NN / SPLIT;           // 2048
    const int n_base0 = split * rows_per_block;
    const float* xb = x + (size_t)b * NN * CC + h * HC;

    for (int chunk = 0; chunk < rows_per_block; chunk += CHUNK) {
        // ---------- stage CHUNK LayerNorm'd rows into LDS ----------
        #pragma unroll
        for (int it = 0; it < CHUNK / 32; ++it) {
            const int r = it * 32 + wave * 4 + rsub;          // 0..CHUNK-1
            const int n = n_base0 + chunk + r;
            const float4 xv = *(const float4*)(xb + (size_t)n * CC + d0);

            float s = xv.x + xv.y + xv.z + xv.w;
            float q = xv.x * xv.x + xv.y * xv.y + xv.z * xv.z + xv.w * xv.w;
            // reduce across the 8 lanes sharing this row (bits 0..2 of lane)
            s += __shfl_xor(s, 1, 32); q += __shfl_xor(q, 1, 32);
            s += __shfl_xor(s, 2, 32); q += __shfl_xor(q, 2, 32);
            s += __shfl_xor(s, 4, 32); q += __shfl_xor(q, 4, 32);

            const float mean = s * (1.0f / 32.0f);
            float var = (q - s * s * (1.0f / 32.0f)) * (1.0f / 31.0f); // unbiased
            var = var > 0.0f ? var : 0.0f;
            const float inv = 1.0f / (sqrtf(var) + EPSF);              // eps on std

            float4 c;
            c.x = (xv.x - mean) * inv; c.y = (xv.y - mean) * inv;
            c.z = (xv.z - mean) * inv; c.w = (xv.w - mean) * inv;

            float4 kk, vv;
            kk.x = c.x * w_k.x + b_k.x; kk.y = c.y * w_k.y + b_k.y;
            kk.z = c.z * w_k.z + b_k.z; kk.w = c.w * w_k.w + b_k.w;
            vv.x = c.x * w_v.x + b_v.x; vv.y = c.y * w_v.y + b_v.y;
            vv.z = c.z * w_v.z + b_v.z; vv.w = c.w * w_v.w + b_v.w;

            *(float4*)(kn_lds + r * PAD + d0) = kk;
            *(float4*)(vn_lds + r * PAD + d0) = vv;
        }
        __syncthreads();

        // ---------- kv tile accumulation via V_WMMA_F32_16X16X4_F32 ----------
        // K-steps (groups of 4 rows) strided across the 8 waves.
        for (int ks = wave; ks < CHUNK / 4; ks += 8) {
            const int nl = ks * 4 + k2;
            v2f a0, a16, bv0, bv16;
            // A tile (d-major): a[m][kk] = k^[n][d0+m]
            a0.x  = kn_lds[(nl + 0) * PAD + m];
            a0.y  = kn_lds[(nl + 1) * PAD + m];
            a16.x = kn_lds[(nl + 0) * PAD + 16 + m];
            a16.y = kn_lds[(nl + 1) * PAD + 16 + m];
            // B tile: b[kk][e] = v^[n][e0+e]
            bv0.x  = vn_lds[(nl + 0) * PAD + m];
            bv0.y  = vn_lds[(nl + 1) * PAD + m];
            bv16.x = vn_lds[(nl + 0) * PAD + 16 + m];
            bv16.y = vn_lds[(nl + 1) * PAD + 16 + m];

            acc[0] = wmma_f32_4(a0,  bv0,  acc[0]);
            acc[1] = wmma_f32_4(a0,  bv16, acc[1]);
            acc[2] = wmma_f32_4(a16, bv0,  acc[2]);
            acc[3] = wmma_f32_4(a16, bv16, acc[3]);
        }
        __syncthreads();
    }

    // ---------- reduce partial kv into global scratch ----------
    float* kvp = kv + (size_t)bh * HC * HC;
    const int mrow = (lane >> 4) * 8;
    #pragma unroll
    for (int t = 0; t < 4; ++t) {
        const int D0 = (t >> 1) * 16;
        const int E0 = (t & 1) * 16;
        #pragma unroll
        for (int r = 0; r < 8; ++r) {
            atomicAdd(kvp + (size_t)(D0 + mrow + r) * HC + E0 + m, acc[t][r]);
        }
    }
}

// ---------------------------------------------------------------------------
// Kernel 2: out = q @ (kv/N) + x    (q = raw x)
// grid = B*H*SPLIT blocks of 256 threads (8 waves); each wave owns 16-row slabs
// ---------------------------------------------------------------------------
__global__ __launch_bounds__(256) void la_pass2(
    const float* __restrict__ x,
    const float* __restrict__ kv,
    float* __restrict__ out)
{
    __shared__ float kvs[HC * HC];

    const int split = blockIdx.x % SPLIT;
    const int bh    = blockIdx.x / SPLIT;
    const int h     = bh % HH;
    const int b     = bh / HH;

    const int tid  = threadIdx.x;
    const int wave = tid >> 5;
    const int lane = tid & 31;

    // stage kv (1024 floats) into LDS
    {
        const float* kvg = kv + (size_t)bh * HC * HC;
        for (int i = tid; i < HC * HC; i += 256) kvs[i] = kvg[i];
    }
    __syncthreads();

    const int m  = lane & 15;
    const int k2 = (lane >> 4) * 2;

    // Preload B operands (kv/N) into registers: 8 K-steps x 2 e-tiles
    v2f bmat[8][2];
    #pragma unroll
    for (int s = 0; s < 8; ++s) {
        const int kk = s * 4 + k2;
        #pragma unroll
        for (int t = 0; t < 2; ++t) {
            const int E0 = t * 16;
            v2f bbv;
            bbv.x = kvs[(kk + 0) * HC + E0 + m] * INVN;
            bbv.y = kvs[(kk + 1) * HC + E0 + m] * INVN;
            bmat[s][t] = bbv;
        }
    }

    const int rows_per_block = NN / SPLIT;     // 2048
    const int nb0 = split * rows_per_block;
    const float* xb = x   + (size_t)b * NN * CC + h * HC;
    float*       ob = out + (size_t)b * NN * CC + h * HC;

    for (int g = 0; g < rows_per_block; g += 128) {
        const int n0 = nb0 + g + wave * 16;
        v8f acc0 = {}; v8f acc1 = {};

        const float* xr = xb + (size_t)(n0 + m) * CC;
        #pragma unroll
        for (int s = 0; s < 8; ++s) {
            const float2 av = *(const float2*)(xr + s * 4 + k2);
            v2f a; a.x = av.x; a.y = av.y;
            acc0 = wmma_f32_4(a, bmat[s][0], acc0);
            acc1 = wmma_f32_4(a, bmat[s][1], acc1);
        }

        // Epilogue: fused residual + store (half-wave writes 64B runs)
        const int rbase = (lane >> 4) * 8;
        #pragma unroll
        for (int r = 0; r < 8; ++r) {
            const size_t i0 = (size_t)(n0 + rbase + r) * CC;
            ob[i0 + m]      = acc0[r] + xb[i0 + m];
            ob[i0 + 16 + m] = acc1[r] + xb[i0 + 16 + m];
        }
    }
}

// ---------------------------------------------------------------------------
extern "C" void kernel_launch(void* const* d_in, const int* in_sizes, int n_in,
                              void* d_out, int out_size, void* d_ws, size_t ws_size,
                              hipStream_t stream)
{
    const float* x  = (const float*)d_in[0];
    const float* wk = (const float*)d_in[1];
    const float* bk = (const float*)d_in[2];
    const float* wv = (const float*)d_in[3];
    const float* bv = (const float*)d_in[4];
    float* out = (float*)d_out;
    float* kv  = (float*)d_ws;          // B*H*32*32 floats = 128 KiB scratch

    (void)in_sizes; (void)n_in; (void)out_size; (void)ws_size;

    const int kv_elems = BB * HH * HC * HC;
    la_zero_kv<<<(kv_elems + 255) / 256, 256, 0, stream>>>(kv);

    const dim3 grid(BB * HH * SPLIT);   // 512 blocks
    la_pass1<<<grid, 256, 0, stream>>>(x, wk, bk, wv, bv, kv);
    la_pass2<<<grid, 256, 0, stream>>>(x, kv, out);
}